// RBF_70042326663914
// MI455X (gfx1250) — compile-verified
//
#include <hip/hip_runtime.h>
#include <math.h>

typedef float v2f __attribute__((ext_vector_type(2)));
typedef float v4f __attribute__((ext_vector_type(4)));
typedef float v8f __attribute__((ext_vector_type(8)));
typedef int   v4i __attribute__((ext_vector_type(4)));

#define DIM   8
#define TASKS 8

__device__ __forceinline__ float softplus_f(float v) {
    // numerically stable softplus: log(1+e^v)
    return (v > 20.0f) ? v : log1pf(__expf(v));
}

// ---------------------------------------------------------------------------
// Kernel 1: tiny precompute — diagonal inverse scales and diagonal variance.
//   inv_s[t*DIM+d] = 1 / softplus(scale_raw[t,t,d])
//   vdiag[t]       = softplus(variance_raw[t,t])
// ---------------------------------------------------------------------------
__global__ void rbf_precompute_params(const float* __restrict__ scale_raw,
                                      const float* __restrict__ variance_raw,
                                      float* __restrict__ inv_s,
                                      float* __restrict__ vdiag) {
    int t = threadIdx.x;
    if (t < TASKS * DIM) {
        int task = t / DIM, d = t % DIM;
        float s = softplus_f(scale_raw[(task * TASKS + task) * DIM + d]);
        inv_s[t] = 1.0f / s;
    }
    if (t < TASKS) {
        vdiag[t] = softplus_f(variance_raw[t * TASKS + t]);
    }
}

// ---------------------------------------------------------------------------
// Kernel 2: per-row scaling by own task's diagonal scale + row metadata.
//   a[n,d]  = x[n,d] * inv_s[idx[n], d]
//   a2[n]   = sum_d a[n,d]^2
//   va[n]   = vdiag[idx[n]]          (pre-gathered variance per row)
// ---------------------------------------------------------------------------
__global__ void rbf_scale_rows(const float* __restrict__ x,
                               const int* __restrict__ idx,
                               const float* __restrict__ inv_s,
                               const float* __restrict__ vdiag,
                               float* __restrict__ a,
                               float* __restrict__ a2,
                               float* __restrict__ va,
                               int nrows) {
    int n = blockIdx.x * blockDim.x + threadIdx.x;
    if (n >= nrows) return;
    int t = idx[n];
    const v4f* xr = (const v4f*)(x + (size_t)n * DIM);
    const v4f* sr = (const v4f*)(inv_s + t * DIM);
    v4f lo = xr[0] * sr[0];
    v4f hi = xr[1] * sr[1];
    ((v4f*)(a + (size_t)n * DIM))[0] = lo;
    ((v4f*)(a + (size_t)n * DIM))[1] = hi;
    float acc = lo.x * lo.x + lo.y * lo.y + lo.z * lo.z + lo.w * lo.w +
                hi.x * hi.x + hi.y * hi.y + hi.z * hi.z + hi.w * hi.w;
    a2[n] = acc;
    va[n] = vdiag[t];
}

// ---------------------------------------------------------------------------
// Kernel 3: one wave32 per 16x16 output tile.
// Cross term a·b via two chained V_WMMA_F32_16X16X4_F32 (K=0..3, K=4..7),
// then branchless masked-exp epilogue with vectorized metadata loads and
// coalesced row-walking stores.
// Grid: x = ceil(tilesM/8) blocks of 8 waves, y = tilesN.
// ---------------------------------------------------------------------------
__global__ void __launch_bounds__(256)
rbf_wmma_kernel(const float* __restrict__ a,   // [Nr*DIM] scaled x rows
                const float* __restrict__ b,   // [Mr*DIM] scaled xx rows
                const float* __restrict__ a2,  // [Nr]
                const float* __restrict__ b2,  // [Mr]
                const int* __restrict__ i_task,   // [Nr]
                const int* __restrict__ ii_task,  // [Mr]
                const float* __restrict__ va,     // [Nr] vdiag[i[n]]
                float* __restrict__ out,
                int Mr) {
    const int lane = threadIdx.x & 31;
    const int wave = threadIdx.x >> 5;
    const int tilesM = Mr >> 4;

    const int tm = blockIdx.x * 8 + wave;
    if (tm >= tilesM) return;  // wave-uniform exit: EXEC all-ones past here
    const int n0 = blockIdx.y << 4;
    const int m0 = tm << 4;

    // ISA 32-bit 16x4 A/B fragment layout (wave32):
    // lane L holds row (L%16); VGPR r holds K = r + (L<16 ? 0 : 2).
    const int lrow  = lane & 15;
    const int khalf = (lane >> 4) << 1;  // 0 or 2

    const float* ap = a + (size_t)(n0 + lrow) * DIM + khalf;
    const float* bp = b + (size_t)(m0 + lrow) * DIM + khalf;

    v2f a_lo = { ap[0], ap[1] };          // K = khalf, khalf+1
    v2f a_hi = { ap[4], ap[5] };          // K = 4+khalf, 4+khalf+1
    v2f b_lo = { bp[0], bp[1] };
    v2f b_hi = { bp[4], bp[5] };

    v8f c = {0.f, 0.f, 0.f, 0.f, 0.f, 0.f, 0.f, 0.f};
    // D = A x B + C ; two K=4 steps cover DIM=8 exactly, full f32 precision
    c = __builtin_amdgcn_wmma_f32_16x16x4_f32(false, a_lo, false, b_lo,
                                              (short)0, c, false, false);
    c = __builtin_amdgcn_wmma_f32_16x16x4_f32(false, a_hi, false, b_hi,
                                              (short)0, c, false, false);

    // C/D layout: lane L -> N = L%16 ; VGPR r -> M = r + (L<16 ? 0 : 8)
    const int m     = m0 + lrow;
    const int nbase = n0 + ((lane >> 4) << 3);   // multiple of 8 -> 32B aligned

    // Per-lane column metadata (one scalar each)
    const float B2m = b2[m];
    const int   tii = ii_task[m];

    // Per-lane row metadata: 8 contiguous rows -> vectorized b128 loads
    const v4f a2lo = *(const v4f*)(a2 + nbase);
    const v4f a2hi = *(const v4f*)(a2 + nbase + 4);
    const v4f valo = *(const v4f*)(va + nbase);
    const v4f vahi = *(const v4f*)(va + nbase + 4);
    const v4i ttlo = *(const v4i*)(i_task + nbase);
    const v4i tthi = *(const v4i*)(i_task + nbase + 4);

    float a2v[8] = {a2lo.x, a2lo.y, a2lo.z, a2lo.w, a2hi.x, a2hi.y, a2hi.z, a2hi.w};
    float vav[8] = {valo.x, valo.y, valo.z, valo.w, vahi.x, vahi.y, vahi.z, vahi.w};
    int   tiv[8] = {ttlo.x, ttlo.y, ttlo.z, ttlo.w, tthi.x, tthi.y, tthi.z, tthi.w};

    const float NEG_INF = -__builtin_huge_valf();
    float* p = out + (size_t)nbase * Mr + m;

#pragma unroll
    for (int r = 0; r < 8; ++r) {
        float d2  = a2v[r] + B2m - 2.0f * c[r];
        // branchless mask: select exponent argument; exp(-inf) == 0
        float arg = (tiv[r] == tii) ? (-0.5f * d2) : NEG_INF;
        float val = __expf(arg) * vav[r];
        *p = val;
        p += Mr;
    }
}

// ---------------------------------------------------------------------------
extern "C" void kernel_launch(void* const* d_in, const int* in_sizes, int n_in,
                              void* d_out, int out_size, void* d_ws, size_t ws_size,
                              hipStream_t stream) {
    const float* x            = (const float*)d_in[0];
    const float* xx           = (const float*)d_in[1];
    const float* scale_raw    = (const float*)d_in[2];
    const float* variance_raw = (const float*)d_in[3];
    const int*   i_task       = (const int*)d_in[4];
    const int*   ii_task      = (const int*)d_in[5];

    const int Nr = in_sizes[0] / DIM;
    const int Mr = in_sizes[1] / DIM;

    // workspace layout (floats); all sections stay 16B-aligned for Nr,Mr mult of 16
    float* ws    = (float*)d_ws;
    float* a     = ws;
    float* b     = a  + (size_t)Nr * DIM;
    float* a2    = b  + (size_t)Mr * DIM;
    float* b2    = a2 + Nr;
    float* va    = b2 + Mr;
    float* vb    = va + Nr;                 // computed but unused in hot kernel
    float* inv_s = vb + Mr;
    float* vdiag = inv_s + TASKS * DIM;

    rbf_precompute_params<<<1, 64, 0, stream>>>(scale_raw, variance_raw, inv_s, vdiag);
    rbf_scale_rows<<<(Nr + 255) / 256, 256, 0, stream>>>(x,  i_task,  inv_s, vdiag,
                                                         a, a2, va, Nr);
    rbf_scale_rows<<<(Mr + 255) / 256, 256, 0, stream>>>(xx, ii_task, inv_s, vdiag,
                                                         b, b2, vb, Mr);

    const int tilesN = Nr >> 4;
    const int tilesM = Mr >> 4;
    dim3 grid((tilesM + 7) / 8, tilesN, 1);  // 8 wave32 tiles per 256-thread block
    rbf_wmma_kernel<<<grid, 256, 0, stream>>>(a, b, a2, b2, i_task, ii_task,
                                              va, (float*)d_out, Mr);
}